// AttLayer_32607391711402
// MI455X (gfx1250) — compile-verified
//
#include <hip/hip_runtime.h>

typedef float v2f __attribute__((ext_vector_type(2)));
typedef float v8f __attribute__((ext_vector_type(8)));

// CDNA5 fp32 matrix op: D(16x16,f32) = A(16x4,f32) x B(4x16,f32) + C
#define WMMA_F32(a, b, c) \
  __builtin_amdgcn_wmma_f32_16x16x4_f32(false, (a), false, (b), (short)0, (c), false, false)

static constexpr int B_  = 8;      // batch
static constexpr int C_  = 256;    // channels
static constexpr int CC_ = 64;     // c = C/4
static constexpr int L_  = 16384;  // sequence
static constexpr int BL_ = 64;     // block length
static constexpr int NB_ = L_ / BL_;

// ---------------------------------------------------------------------------
// 1x1 conv as GEMM: Y[b][m][l] = sum_ch W[m][ch] * X[b][ch][l] + bias
// DUAL: also Y2 = W2*X + bias2 in the same pass (shares all X loads).
// Block tile 64(M) x 64(N); 8 waves: m-tile (w&3), n-tiles (w>>2)*16 / +32.
// W^T staged per K-chunk in LDS, pair-interleaved -> A fragment is one b64 DS
// load. Chunking keeps static LDS <= 64KB even in DUAL mode.
// ---------------------------------------------------------------------------
template <int CIN, bool DUAL, bool MASKED>
__global__ __launch_bounds__(256) void conv1x1_wmma(
    const float* __restrict__ X,
    const float* __restrict__ W,  const float* __restrict__ bias,
    const float* __restrict__ W2, const float* __restrict__ bias2,
    const float* __restrict__ mask,
    float* __restrict__ Y, float* __restrict__ Y2, int CoutT) {
  constexpr int CHUNK = (CIN >= 128) ? 128 : CIN;
  __shared__ float ldsW[(DUAL ? 2 : 1) * CHUNK * 64];  // [(k>>1)][m][k&1]

  const int tid   = threadIdx.x;
  const int b     = blockIdx.y;
  const int mz    = blockIdx.z;          // 64-row M slice
  const int nbase = blockIdx.x * 64;

  const int wave = tid >> 5, lane = tid & 31;
  const int half = lane >> 4, mr = lane & 15;
  const int m0 = (wave & 3) * 16;
  const int n0 = (wave >> 2) * 16;
  const int col0 = n0 + mr, col1 = n0 + 32 + mr;
  const float* Xb = X + (size_t)b * CIN * L_ + nbase;

  v8f accA0 = {}, accA1 = {}, accB0 = {}, accB1 = {};
  const v2f* ldsA  = (const v2f*)ldsW;
  const v2f* ldsA2 = (const v2f*)(ldsW + CHUNK * 64);

  for (int k0 = 0; k0 < CIN; k0 += CHUNK) {
    __syncthreads();  // previous chunk fully consumed
    for (int i = tid; i < 64 * CHUNK; i += 256) {  // uniform trip count
      int mm = i / CHUNK, chl = i % CHUNK;
      float w = W[(size_t)(mz * 64 + mm) * CIN + k0 + chl];
      ldsW[(chl >> 1) * 128 + mm * 2 + (chl & 1)] = w;
      if (DUAL) {
        float w2 = W2[(size_t)(mz * 64 + mm) * CIN + k0 + chl];
        ldsW[CHUNK * 64 + (chl >> 1) * 128 + mm * 2 + (chl & 1)] = w2;
      }
    }
    __syncthreads();

    for (int kk = 0; kk < CHUNK / 4; ++kk) {
      const int krow = k0 + 4 * kk + 2 * half;
      v2f a = ldsA[(2 * kk + half) * 64 + m0 + mr];
      v2f b0, b1;
      b0.x = Xb[(size_t)krow * L_ + col0];
      b0.y = Xb[(size_t)(krow + 1) * L_ + col0];
      b1.x = Xb[(size_t)krow * L_ + col1];
      b1.y = Xb[(size_t)(krow + 1) * L_ + col1];
      if (kk + 2 < CHUNK / 4)
        __builtin_prefetch(&Xb[(size_t)(krow + 8) * L_ + col0], 0, 0);
      accA0 = WMMA_F32(a, b0, accA0);
      accA1 = WMMA_F32(a, b1, accA1);
      if (DUAL) {
        v2f a2 = ldsA2[(2 * kk + half) * 64 + m0 + mr];
        accB0 = WMMA_F32(a2, b0, accB0);
        accB1 = WMMA_F32(a2, b1, accB1);
      }
    }
  }

  float mk0 = 1.f, mk1 = 1.f;
  if (MASKED) {
    mk0 = mask[(size_t)b * L_ + nbase + col0];
    mk1 = mask[(size_t)b * L_ + nbase + col1];
  }
  float* Yb  = Y + (size_t)b * CoutT * L_ + nbase;
  float* Y2b = DUAL ? (Y2 + (size_t)b * CoutT * L_ + nbase) : nullptr;
  for (int r = 0; r < 8; ++r) {
    const int mg = mz * 64 + m0 + r + 8 * half;  // C/D row = m0 + r + 8*half
    float v0 = accA0[r] + bias[mg];
    float v1 = accA1[r] + bias[mg];
    if (MASKED) { v0 *= mk0; v1 *= mk1; }
    Yb[(size_t)mg * L_ + col0] = v0;
    Yb[(size_t)mg * L_ + col1] = v1;
    if (DUAL) {
      Y2b[(size_t)mg * L_ + col0] = accB0[r] + bias2[mg];
      Y2b[(size_t)mg * L_ + col1] = accB1[r] + bias2[mg];
    }
  }
}

// ---------------------------------------------------------------------------
// Block-local attention for one (block, batch):
//   energy(64l x 128j) = Q^T Kwin / 8 ; att = softmax(energy + log(fm+1e-6))*fm
//   y(64ch x 64l) = relu( Vwin x att^T )
// Window j -> global column blk*64 - 64 + j. Left padding handled branchlessly:
// invalid lanes read a safe in-range address (+64) and multiply by 0.
// ---------------------------------------------------------------------------
__global__ __launch_bounds__(256) void attn_wmma(
    const float* __restrict__ q, const float* __restrict__ k,
    const float* __restrict__ v, const float* __restrict__ mask,
    float* __restrict__ y) {
  constexpr int P = 132;  // pitch % 64 == 4 -> conflict-free strided b64 DS loads
  __shared__ float e[64 * P];
  __shared__ float mwin[128];

  const int tid = threadIdx.x;
  const int blk = blockIdx.x, b = blockIdx.y;
  const int base = blk * 64;

  if (tid < 128) {
    int cj = base - 64 + tid;
    mwin[tid] = (cj >= 0) ? mask[(size_t)b * L_ + cj] : 0.f;
  }

  const int wave = tid >> 5, lane = tid & 31;
  const int half = lane >> 4, mr = lane & 15;
  const float* qb = q + (size_t)b * CC_ * L_ + base;
  const float* kb = k + (size_t)b * CC_ * L_ + base - 64;  // window origin
  const float* vb = v + (size_t)b * CC_ * L_ + base - 64;

  // ---- energy = Q^T Kwin: 4x8 tiles of 16x16; wave -> lt=(w&3), jg=(w>>2)
  {
    const int l0 = (wave & 3) * 16;
    const int jg = wave >> 2;
    int   jsafe[4];
    float jvf[4];
    for (int t = 0; t < 4; ++t) {
      const int jc = (jg + 2 * t) * 16 + mr;
      const bool val = (blk > 0) || (jc >= 64);
      jsafe[t] = val ? jc : jc + 64;  // always in-range column
      jvf[t]   = val ? 1.f : 0.f;
    }
    v8f acc[4] = {{}, {}, {}, {}};
    for (int kk = 0; kk < 16; ++kk) {           // K = 64 channels
      const int krow = 4 * kk + 2 * half;
      v2f a;                                    // A[m=l][k=ch] = q[ch][l]
      a.x = qb[(size_t)krow * L_ + l0 + mr];
      a.y = qb[(size_t)(krow + 1) * L_ + l0 + mr];
      for (int t = 0; t < 4; ++t) {
        v2f bm;                                 // B[k=ch][n=j] = kwin[ch][j]
        bm.x = kb[(size_t)krow * L_ + jsafe[t]] * jvf[t];
        bm.y = kb[(size_t)(krow + 1) * L_ + jsafe[t]] * jvf[t];
        acc[t] = WMMA_F32(a, bm, acc[t]);
      }
    }
    for (int t = 0; t < 4; ++t) {
      const int j0 = (jg + 2 * t) * 16;
      for (int r = 0; r < 8; ++r)
        e[(l0 + r + 8 * half) * P + j0 + mr] = acc[t][r] * 0.125f;  // /sqrt(64)
    }
  }
  __syncthreads();

  // ---- softmax(energy + log(fm+1e-6)) * fm: 4 lanes per row, shfl reductions
  {
    const int l  = tid >> 2;
    const int jb = (tid & 3) * 32;
    float mx = -3.4e38f;
    for (int jj = 0; jj < 32; ++jj) {
      const int j = jb + jj;
      float fm  = ((j >= l) && (j < l + 64)) ? mwin[j] : 0.f;
      float val = e[l * P + j] + __logf(fm + 1e-6f);
      e[l * P + j] = val;
      mx = fmaxf(mx, val);
    }
    mx = fmaxf(mx, __shfl_xor(mx, 1));
    mx = fmaxf(mx, __shfl_xor(mx, 2));
    float s = 0.f;
    for (int jj = 0; jj < 32; ++jj) {
      const int j = jb + jj;
      float p = __expf(e[l * P + j] - mx);
      e[l * P + j] = p;
      s += p;
    }
    s += __shfl_xor(s, 1);
    s += __shfl_xor(s, 2);
    const float inv = 1.f / s;
    for (int jj = 0; jj < 32; ++jj) {
      const int j = jb + jj;
      float fm = ((j >= l) && (j < l + 64)) ? mwin[j] : 0.f;
      e[l * P + j] *= inv * fm;
    }
  }
  __syncthreads();

  // ---- y = relu(Vwin x att^T): M=64ch, N=64l, K=128j; 4x4 tiles, 2 per wave
  {
    const int m0 = (wave & 3) * 16;   // ch tile
    const int n0 = (wave >> 2) * 16;  // l tiles: n0 and n0+32
    v8f acc0 = {}, acc1 = {};
    const v2f* eA = (const v2f*)e;    // P even, krow even -> aligned pairs
    for (int kk = 0; kk < 32; ++kk) {
      const int krow = 4 * kk + 2 * half;          // j index
      const bool val = (blk > 0) || (krow >= 64);
      const int   ks = val ? krow : krow + 64;     // safe in-range column
      const float vf = val ? 1.f : 0.f;
      v2f a;                                       // A[m=ch][k=j] = vwin[ch][j]
      a.x = vb[(size_t)(m0 + mr) * L_ + ks] * vf;
      a.y = vb[(size_t)(m0 + mr) * L_ + ks + 1] * vf;
      v2f b0 = eA[((n0 + mr) * P + krow) >> 1];    // B[k=j][n=l] = att[l][j]
      v2f b1 = eA[((n0 + 32 + mr) * P + krow) >> 1];
      acc0 = WMMA_F32(a, b0, acc0);
      acc1 = WMMA_F32(a, b1, acc1);
    }
    float* yb = y + (size_t)b * CC_ * L_ + base;
    for (int r = 0; r < 8; ++r) {
      const int ch = m0 + r + 8 * half;
      yb[(size_t)ch * L_ + n0 + mr]      = fmaxf(acc0[r], 0.f);  // relu fused
      yb[(size_t)ch * L_ + n0 + 32 + mr] = fmaxf(acc1[r], 0.f);
    }
  }
}

// ---------------------------------------------------------------------------
extern "C" void kernel_launch(void* const* d_in, const int* in_sizes, int n_in,
                              void* d_out, int out_size, void* d_ws, size_t ws_size,
                              hipStream_t stream) {
  (void)in_sizes; (void)n_in; (void)out_size; (void)ws_size;
  const float* x1   = (const float*)d_in[0];
  const float* x2   = (const float*)d_in[1];
  const float* mask = (const float*)d_in[2];
  const float* Wq   = (const float*)d_in[3];
  const float* bq   = (const float*)d_in[4];
  const float* Wk   = (const float*)d_in[5];
  const float* bk   = (const float*)d_in[6];
  const float* Wv   = (const float*)d_in[7];
  const float* bv   = (const float*)d_in[8];
  const float* Wo   = (const float*)d_in[9];
  const float* bo   = (const float*)d_in[10];
  float* out = (float*)d_out;

  // workspace: q, k, v, y  (each B*c*L f32 = 32 MB; total 128 MB)
  const size_t seg = (size_t)B_ * CC_ * L_;
  float* q  = (float*)d_ws;
  float* kq = q + seg;
  float* vq = kq + seg;
  float* y  = vq + seg;

  dim3 blk(256);
  dim3 gqkv(L_ / 64, B_, 1);
  // q and k fused: one pass over x1 feeds both WMMA chains
  conv1x1_wmma<C_, true, false><<<gqkv, blk, 0, stream>>>(
      x1, Wq, bq, Wk, bk, nullptr, q, kq, CC_);
  conv1x1_wmma<C_, false, false><<<gqkv, blk, 0, stream>>>(
      x2, Wv, bv, nullptr, nullptr, nullptr, vq, nullptr, CC_);

  attn_wmma<<<dim3(NB_, B_), blk, 0, stream>>>(q, kq, vq, mask, y);

  conv1x1_wmma<CC_, false, true><<<dim3(L_ / 64, B_, C_ / 64), blk, 0, stream>>>(
      y, Wo, bo, nullptr, nullptr, mask, out, nullptr, C_);
}